// GIN_16252156248696
// MI455X (gfx1250) — compile-verified
//
#include <hip/hip_runtime.h>
#include <hip/hip_bf16.h>
#include <math.h>

typedef __attribute__((ext_vector_type(2))) float v2f;
typedef __attribute__((ext_vector_type(8))) float v8f;

#define NUM_GRAPHS 1024

__device__ __forceinline__ void atomic_add_f32(float* p, float v) {
  // agent-scope relaxed FP add -> native global_atomic_add_f32 (non-returning)
  __hip_atomic_fetch_add(p, v, __ATOMIC_RELAXED, __HIP_MEMORY_SCOPE_AGENT);
}

// ---------------- kernel 1: zero workspace ----------------
__global__ void gin_zero_kernel(float* __restrict__ ws, long long n) {
  long long i = (long long)blockIdx.x * blockDim.x + threadIdx.x;
  long long stride = (long long)gridDim.x * blockDim.x;
  for (; i < n; i += stride) ws[i] = 0.0f;
}

// ---------------- kernel 2: agg[dst] += x[src] over all edges ----------------
// edge_index (128 MB) is streamed once -> non-temporal loads keep it from
// evicting the L2-resident x/agg working set (16 MB of 192 MB L2).
__global__ void gin_edge_scatter_kernel(const float4* __restrict__ x4,
                                        const int* __restrict__ src,
                                        const int* __restrict__ dst,
                                        float* __restrict__ agg,
                                        long long E) {
  long long e = (long long)blockIdx.x * blockDim.x + threadIdx.x;
  if (e >= E) return;
  int s = __builtin_nontemporal_load(src + e);
  int d = __builtin_nontemporal_load(dst + e);
  float4 xs = x4[s];                 // global_load_b128, L2-resident gather
  float* a = agg + (long long)d * 4; // L2-resident scatter target
  atomic_add_f32(a + 0, xs.x);
  atomic_add_f32(a + 1, xs.y);
  atomic_add_f32(a + 2, xs.z);
  atomic_add_f32(a + 3, xs.w);
}

// ---------------- kernel 3: (1+eps)*x + agg -> MLP via WMMA -> pooled atomics ----------------
// One wave (32 lanes) processes a tile of 16 nodes.
//   Layer 1: D1[16x16] = A[16 nodes x 4] * W1[4x16]        (1x v_wmma_f32_16x16x4_f32)
//   Layer 2: D2[16x16] = relu(D1+b1)[16x16] * W2pad[16x16] (4x chained v_wmma_f32_16x16x4_f32)
// A-layout (16x4 f32): lane<16 -> row M=lane, K={0,1}; lane>=16 -> row M=lane-16, K={2,3}
// B-layout (4x16 f32): lane<16 -> col N=lane, K={0,1}; lane>=16 -> col N=lane-16, K={2,3}
// C/D-layout: VGPR r -> rows {r, r+8}; cols across lanes (N = lane&15, +8 rows for lane>=16)
__global__ void __launch_bounds__(256)
gin_mlp_pool_kernel(const float* __restrict__ x,
                    const float* __restrict__ agg,
                    const int* __restrict__ batch,
                    const float* __restrict__ epsp,
                    const float* __restrict__ W1, const float* __restrict__ b1,
                    const float* __restrict__ W2, const float* __restrict__ b2,
                    float* __restrict__ sums, float* __restrict__ cnts,
                    int N) {
  __shared__ float lds[8][256];                 // per-wave 16x16 f32 transpose staging
  const int lane = threadIdx.x & 31;
  const int wave = threadIdx.x >> 5;
  const int n    = lane & 15;                   // column / row-in-tile index
  const int hi   = lane >> 4;                   // 0 -> K pair {0,1}, 1 -> K pair {2,3}
  const int tile = blockIdx.x * 8 + wave;
  const int base = tile * 16;
  const float onepe = 1.0f + epsp[0];

  // ----- layer 1 operands -----
  int nodeA  = base + n;
  int nodeAc = nodeA < N ? nodeA : (N - 1);     // clamped (results predicated later)
  const float* xr = x   + (long long)nodeAc * 4 + hi * 2;
  const float* gr = agg + (long long)nodeAc * 4 + hi * 2;
  v2f A; A.x = onepe * xr[0] + gr[0];
         A.y = onepe * xr[1] + gr[1];
  v2f B; B.x = W1[(hi * 2 + 0) * 16 + n];       // W1 row-major [4][16]
         B.y = W1[(hi * 2 + 1) * 16 + n];

  // ----- branch-free preload of layer-2 weights (overlaps barrier latency) -----
  const float mask2 = (n < 4) ? 1.0f : 0.0f;
  const int   nc    = n & 3;
  float w2a[4], w2b[4];
#pragma unroll
  for (int j = 0; j < 4; ++j) {
    const int k0 = j * 4 + hi * 2;
    w2a[j] = W2[(k0 + 0) * 4 + nc] * mask2;     // W2 row-major [16][4], N-padded via mask
    w2b[j] = W2[(k0 + 1) * 4 + nc] * mask2;
  }
  const float bb1 = b1[n];
  const float bb2 = b2[nc];

  v8f C = {};
  C = __builtin_amdgcn_wmma_f32_16x16x4_f32(false, A, false, B, (short)0, C,
                                            false, false);

  // bias + ReLU, write hidden activations [M][K] row-major into LDS
#pragma unroll
  for (int r = 0; r < 8; ++r) {
    float h = C[r] + bb1;
    h = h > 0.0f ? h : 0.0f;
    lds[wave][(r + 8 * hi) * 16 + n] = h;
  }
  __syncthreads();                              // uniform for all waves in block

  // ----- layer 2: 4 chained K=4 WMMAs over hidden dim 16 -----
  v8f C2 = {};
#pragma unroll
  for (int j = 0; j < 4; ++j) {
    const int k0 = j * 4 + hi * 2;
    v2f A2; A2.x = lds[wave][n * 16 + k0 + 0];  // A row M=n, K slice (ds_load)
            A2.y = lds[wave][n * 16 + k0 + 1];
    v2f B2; B2.x = w2a[j];
            B2.y = w2b[j];
    C2 = __builtin_amdgcn_wmma_f32_16x16x4_f32(false, A2, false, B2, (short)0, C2,
                                               false, false);
  }

  // ----- bias + ReLU + per-graph pooled accumulation -----
  // batch is sorted -> run-length-compress the 8 sorted ids this lane covers
  // before hitting global atomics (~8x fewer pooled atomics in the common case).
  const bool fullTile = (base + 16 <= N);       // wave-uniform
  if (fullTile) {
    const int4* bp = (const int4*)(batch + base + 8 * hi);  // 16B-aligned
    int4 q0 = bp[0];
    int4 q1 = bp[1];
    int bid[8] = {q0.x, q0.y, q0.z, q0.w, q1.x, q1.y, q1.z, q1.w};

    if (n < 4) {                                // rows m = 8*hi + r, feature n
      float vv[8];
#pragma unroll
      for (int r = 0; r < 8; ++r) {
        float v = C2[r] + bb2;
        vv[r] = v > 0.0f ? v : 0.0f;
      }
      float acc = vv[0];
      int   cur = bid[0];
#pragma unroll
      for (int r = 1; r < 8; ++r) {
        if (bid[r] == cur) {
          acc += vv[r];
        } else {
          atomic_add_f32(&sums[(long long)cur * 4 + n], acc);
          cur = bid[r];
          acc = vv[r];
        }
      }
      atomic_add_f32(&sums[(long long)cur * 4 + n], acc);
    }
    if (n == 0) {                               // one lane per half-tile: counts
      float acc = 1.0f;
      int   cur = bid[0];
#pragma unroll
      for (int r = 1; r < 8; ++r) {
        if (bid[r] == cur) {
          acc += 1.0f;
        } else {
          atomic_add_f32(&cnts[cur], acc);
          cur = bid[r];
          acc = 1.0f;
        }
      }
      atomic_add_f32(&cnts[cur], acc);
    }
  } else {                                      // partial tail tile (rare)
    if (n < 4) {
#pragma unroll
      for (int r = 0; r < 8; ++r) {
        int node = base + r + 8 * hi;
        if (node < N) {
          float v = C2[r] + bb2;
          v = v > 0.0f ? v : 0.0f;
          atomic_add_f32(&sums[(long long)batch[node] * 4 + n], v);
        }
      }
    }
    if (hi == 0) {
      int node = base + n;
      if (node < N) atomic_add_f32(&cnts[batch[node]], 1.0f);
    }
  }
}

// ---------------- kernel 4: mean pool + log_softmax ----------------
__global__ void gin_pool_lsm_kernel(const float* __restrict__ sums,
                                    const float* __restrict__ cnts,
                                    float* __restrict__ out) {
  int g = blockIdx.x * blockDim.x + threadIdx.x;
  if (g >= NUM_GRAPHS) return;
  float c = cnts[g];
  c = c > 1.0f ? c : 1.0f;
  float v0 = sums[g * 4 + 0] / c;
  float v1 = sums[g * 4 + 1] / c;
  float v2 = sums[g * 4 + 2] / c;
  float v3 = sums[g * 4 + 3] / c;
  float m = fmaxf(fmaxf(v0, v1), fmaxf(v2, v3));
  float s = __expf(v0 - m) + __expf(v1 - m) + __expf(v2 - m) + __expf(v3 - m);
  float lse = m + __logf(s);
  out[g * 4 + 0] = v0 - lse;
  out[g * 4 + 1] = v1 - lse;
  out[g * 4 + 2] = v2 - lse;
  out[g * 4 + 3] = v3 - lse;
}

extern "C" void kernel_launch(void* const* d_in, const int* in_sizes, int n_in,
                              void* d_out, int out_size, void* d_ws, size_t ws_size,
                              hipStream_t stream) {
  const float* x          = (const float*)d_in[0];
  const int*   edge_index = (const int*)d_in[1];
  const int*   batch      = (const int*)d_in[2];
  const float* eps        = (const float*)d_in[3];
  const float* W1         = (const float*)d_in[4];
  const float* b1         = (const float*)d_in[5];
  const float* W2         = (const float*)d_in[6];
  const float* b2         = (const float*)d_in[7];

  const int       N = in_sizes[0] / 4;
  const long long E = (long long)in_sizes[1] / 2;
  const int*   src  = edge_index;          // edge_index[0][:]
  const int*   dst  = edge_index + E;      // edge_index[1][:]

  // workspace layout: agg[N*4] | sums[G*4] | cnts[G]
  float* agg  = (float*)d_ws;
  float* sums = agg + (size_t)N * 4;
  float* cnts = sums + (size_t)NUM_GRAPHS * 4;

  // 1) zero accumulators
  long long zn = (long long)N * 4 + NUM_GRAPHS * 4 + NUM_GRAPHS;
  int zblocks = (int)((zn + 255) / 256);
  if (zblocks > 4096) zblocks = 4096;
  gin_zero_kernel<<<zblocks, 256, 0, stream>>>((float*)d_ws, zn);

  // 2) edge scatter-add (dominant cost: ~640 MB of L2-resident traffic)
  int eblocks = (int)((E + 255) / 256);
  gin_edge_scatter_kernel<<<eblocks, 256, 0, stream>>>(
      (const float4*)x, src, dst, agg, E);

  // 3) fused GIN update + WMMA MLP + pooled accumulation (8 waves/block, 1 tile/wave)
  int numTiles = (N + 15) / 16;
  int mblocks  = (numTiles + 7) / 8;
  gin_mlp_pool_kernel<<<mblocks, 256, 0, stream>>>(
      x, agg, batch, eps, W1, b1, W2, b2, sums, cnts, N);

  // 4) mean + log_softmax
  gin_pool_lsm_kernel<<<(NUM_GRAPHS + 255) / 256, 256, 0, stream>>>(
      sums, cnts, (float*)d_out);
}